// AdaptiveVoxelization_30528627540701
// MI455X (gfx1250) — compile-verified
//
#include <hip/hip_runtime.h>

// ---------------------------------------------------------------------------
// AdaptiveVoxelization for MI455X (gfx1250, wave32)
//   out[b, off(res)+flat, :] = scale[b,res] * sum_{points in voxel} point
// Strategy (verified lowering via compile probe):
//   res 8/16  -> LDS accumulation (ds_add_f32), one atomic flush per WG
//   res 32/64 -> direct global_atomic_add_f32 (output is L2-resident: 115MB<192MB)
//   point data -> triple-buffered async global->LDS staging
//                 (global_load_async_to_lds_b32 + s_wait_asynccnt)
//   scales fused into per-point contributions -> single pass
// ---------------------------------------------------------------------------

typedef float v4f __attribute__((ext_vector_type(4)));

constexpr int B_         = 32;
constexpr int N_         = 200000;
constexpr int CHUNKS     = 16;                    // chunks per batch
constexpr int CHUNK_PTS  = N_ / CHUNKS;           // 12500
constexpr int CHUNK_BYTES= CHUNK_PTS * 12;        // 150000
constexpr int TILE       = 256;                   // points per tile (== blockDim)
constexpr int NTILES     = (CHUNK_PTS + TILE - 1) / TILE;  // 49
constexpr int ROWS       = 512 + 4096 + 32768 + 262144;    // 299520
constexpr int OFF16      = 512;
constexpr int OFF32      = 512 + 4096;            // 4608
constexpr int OFF64      = 512 + 4096 + 32768;    // 37376

#define ASYNC_OK (__has_builtin(__builtin_amdgcn_global_load_async_to_lds_b32) && \
                  __has_builtin(__builtin_amdgcn_s_wait_asynccnt))

#if defined(__HIP_DEVICE_COMPILE__)
typedef __attribute__((address_space(1))) int* gas_i32p;
typedef __attribute__((address_space(3))) int* las_i32p;
#endif

__device__ __forceinline__ int vidx(float v, float rf, int rm1) {
  int i = (int)(v * rf);                 // trunc == floor for v >= 0
  i = i > rm1 ? rm1 : i;                 // guard fp round-up at v -> 1.0-
  return i < 0 ? 0 : i;
}

__device__ __forceinline__ void fatomic(float* p, float v) {
  (void)unsafeAtomicAdd(p, v);           // global -> global_atomic_add_f32 (verified)
}                                        // shared -> ds_add_f32 (verified)

__global__ __launch_bounds__(256) void vox_scatter_kernel(
    const float* __restrict__ points, const float* __restrict__ rmap,
    float* __restrict__ out) {
  __shared__ float g8[512 * 3];            //  6 KB, scaled res8 accumulator
  __shared__ float g16[4096 * 3];          // 48 KB, scaled res16 accumulator
  __shared__ float pbuf[3][TILE * 3];      //  9 KB, triple-buffered point stage

  const int tid   = threadIdx.x;
  const int chunk = blockIdx.x;
  const int b     = blockIdx.y;

  for (int i = tid; i < 512 * 3;  i += TILE) g8[i]  = 0.f;
  for (int i = tid; i < 4096 * 3; i += TILE) g16[i] = 0.f;

  const float s8  = rmap[b * 4 + 0];
  const float s16 = rmap[b * 4 + 1];
  const float s32 = rmap[b * 4 + 2];
  const float s64 = rmap[b * 4 + 3];

  const char* gbase =
      (const char*)(points + ((size_t)b * N_ + (size_t)chunk * CHUNK_PTS) * 3);
  const unsigned obase = (unsigned)b * (unsigned)(ROWS * 3);  // 32-bit offsets

  // Stage one tile (3072 B) of points into pbuf[buf], coalesced b32 per lane.
  auto issueTile = [&](int t, int buf) {
    const int base = t * (TILE * 12);
#pragma unroll
    for (int j = 0; j < 3; ++j) {
      const int foff = tid + j * TILE;                 // float slot 0..767
      int goff = base + foff * 4;
      if (goff > CHUNK_BYTES - 4) goff = CHUNK_BYTES - 4;  // clamp: stay in-bounds
#if defined(__HIP_DEVICE_COMPILE__) && ASYNC_OK
      __builtin_amdgcn_global_load_async_to_lds_b32(
          (gas_i32p)(gbase + goff),
          (las_i32p)&pbuf[buf][foff],
          0, 0);
#else
      pbuf[buf][foff] = *(const float*)(gbase + goff);
#endif
    }
  };

  issueTile(0, 0);                                     // prologue

  for (int t = 0; t < NTILES; ++t) {
    if (t + 1 < NTILES) {
      issueTile(t + 1, (t + 1) % 3);                   // overlap next tile's DMA
#if defined(__HIP_DEVICE_COMPILE__) && ASYNC_OK
      __builtin_amdgcn_s_wait_asynccnt(3);             // in-order: tile t retired
#endif
    } else {
#if defined(__HIP_DEVICE_COMPILE__) && ASYNC_OK
      __builtin_amdgcn_s_wait_asynccnt(0);
#endif
    }
    __syncthreads();                                   // tile t visible to all waves

    const int p = t * TILE + tid;
    if (p < CHUNK_PTS) {
      const float* q = &pbuf[t % 3][tid * 3];
      const float x = q[0], y = q[1], z = q[2];

      // res 8 -> LDS
      const int i8 = (vidx(x, 8.f, 7) * 8 + vidx(y, 8.f, 7)) * 8 + vidx(z, 8.f, 7);
      fatomic(&g8[i8 * 3 + 0], x * s8);
      fatomic(&g8[i8 * 3 + 1], y * s8);
      fatomic(&g8[i8 * 3 + 2], z * s8);

      // res 16 -> LDS
      const int i16 =
          (vidx(x, 16.f, 15) * 16 + vidx(y, 16.f, 15)) * 16 + vidx(z, 16.f, 15);
      fatomic(&g16[i16 * 3 + 0], x * s16);
      fatomic(&g16[i16 * 3 + 1], y * s16);
      fatomic(&g16[i16 * 3 + 2], z * s16);

      // res 32 -> global (L2-resident), 32-bit offsets
      const int i32 =
          (vidx(x, 32.f, 31) * 32 + vidx(y, 32.f, 31)) * 32 + vidx(z, 32.f, 31);
      const unsigned o32 = obase + (unsigned)(OFF32 + i32) * 3u;
      fatomic(out + o32 + 0, x * s32);
      fatomic(out + o32 + 1, y * s32);
      fatomic(out + o32 + 2, z * s32);

      // res 64 -> global (L2-resident), 32-bit offsets
      const int i64 =
          (vidx(x, 64.f, 63) * 64 + vidx(y, 64.f, 63)) * 64 + vidx(z, 64.f, 63);
      const unsigned o64 = obase + (unsigned)(OFF64 + i64) * 3u;
      fatomic(out + o64 + 0, x * s64);
      fatomic(out + o64 + 1, y * s64);
      fatomic(out + o64 + 2, z * s64);
    }
  }

  // Flush LDS accumulators to global with one atomic pass (skip zeros).
  __syncthreads();
  for (int i = tid; i < 512 * 3; i += TILE) {
    const float v = g8[i];
    if (v != 0.f) fatomic(out + obase + i, v);         // res8 offset == 0
  }
  for (int i = tid; i < 4096 * 3; i += TILE) {
    const float v = g16[i];
    if (v != 0.f) fatomic(out + obase + OFF16 * 3 + i, v);
  }
}

__global__ __launch_bounds__(256) void vox_zero_kernel(v4f* __restrict__ out, int n4) {
  int i      = blockIdx.x * blockDim.x + threadIdx.x;
  int stride = gridDim.x * blockDim.x;
  v4f z = {0.f, 0.f, 0.f, 0.f};
  for (; i < n4; i += stride) out[i] = z;   // keep RT hint: want lines in L2 for atomics
}

extern "C" void kernel_launch(void* const* d_in, const int* in_sizes, int n_in,
                              void* d_out, int out_size, void* d_ws, size_t ws_size,
                              hipStream_t stream) {
  const float* points = (const float*)d_in[0];   // [32, 200000, 3] f32
  const float* rmap   = (const float*)d_in[1];   // [32, 4, 1] f32
  float* out          = (float*)d_out;           // [32, 299520, 3] f32

  const int n4 = out_size / 4;                   // 7,188,480 float4s (16B aligned)
  vox_zero_kernel<<<dim3(8192), dim3(256), 0, stream>>>((v4f*)out, n4);
  vox_scatter_kernel<<<dim3(CHUNKS, B_), dim3(256), 0, stream>>>(points, rmap, out);
}